// RNN_11081015623879
// MI455X (gfx1250) — compile-verified
//
#include <hip/hip_runtime.h>
#include <math.h>

typedef __attribute__((ext_vector_type(2))) float v2f;
typedef __attribute__((ext_vector_type(8))) float v8f;

#define HID  4096
#define NCAT 1024
#define INP  4096
#define KXI  (NCAT + INP)   // 5120
#define OUTN 50257

__device__ __forceinline__ float sigmoid_f(float v) {
    return 1.0f / (1.0f + __expf(-v));
}

// Build x = [category, input]; also seed oc[0:NCAT] = category.
__global__ void build_x_kernel(const float* __restrict__ cat,
                               const float* __restrict__ inp,
                               float* __restrict__ x,
                               float* __restrict__ oc) {
    int i = blockIdx.x * blockDim.x + threadIdx.x;
    if (i < NCAT) {
        float c = cat[i];
        x[i]  = c;
        oc[i] = c;
    } else if (i < KXI) {
        x[i] = inp[i - NCAT];
    }
}

// out[j] = dot(vec[0:K], W[j][0:K]) + bias[j]  for j in [0, rows)
// One wave handles 16 consecutive output rows via V_WMMA_F32_16X16X4_F32:
//   B tile (4x16): lane L streams W row j0+(L&15), K-pair offset 2*(L>>4)
//   A tile row 0 : the vector (only D row 0 is consumed)
// 256 threads = 8 waves = 128 rows per block.
__global__ __launch_bounds__(256) void wmma_matvec_kernel(
    const float* __restrict__ W, const float* __restrict__ vec,
    const float* __restrict__ bias, float* __restrict__ out,
    int rows, int K)
{
    const int lane = threadIdx.x & 31;
    const int wave = threadIdx.x >> 5;
    const int j0   = (blockIdx.x * 8 + wave) * 16;
    if (j0 >= rows) return;              // wave-uniform: EXEC stays all-ones

    int r = j0 + (lane & 15);
    if (r > rows - 1) r = rows - 1;      // tail rows: duplicate loads, guarded store
    const int kh = (lane >> 4) * 2;      // 0 for lanes 0-15, 2 for lanes 16-31

    const float* __restrict__ wp = W + (size_t)r * (size_t)K + kh;
    const float* __restrict__ vp = vec + kh;

    float acc;
#if __has_builtin(__builtin_amdgcn_wmma_f32_16x16x4_f32)
    v8f c = {};
    for (int k0 = 0; k0 < K; k0 += 32) {
#pragma unroll
        for (int u = 0; u < 8; ++u) {
            const int k = k0 + u * 4;
            v2f b = *(const v2f*)(wp + k);   // global_load_b64: weight pair
            v2f a = *(const v2f*)(vp + k);   // vector pair (L0-resident, tiny)
            // 8 args: (neg_a, A, neg_b, B, c_mod, C, reuse_a, reuse_b)
            c = __builtin_amdgcn_wmma_f32_16x16x4_f32(
                    false, a, false, b, (short)0, c, false, false);
        }
    }
    acc = c[0];   // lanes 0..15 hold D[0][n=lane]
#else
    acc = 0.0f;
    if (lane < 16) {
        const float* __restrict__ wr = W + (size_t)r * (size_t)K;
        for (int k = 0; k < K; ++k) acc = fmaf(vec[k], wr[k], acc);
    }
#endif

    if (lane < 16) {
        const int j = j0 + lane;
        if (j < rows) out[j] = acc + bias[j];
    }
}

// GRU gate math (PyTorch gate order r,z,n); writes h_new to output and into oc.
__global__ void gru_gate_kernel(const float* __restrict__ gi,
                                const float* __restrict__ gh,
                                const float* __restrict__ hidden,
                                float* __restrict__ h_out,
                                float* __restrict__ oc) {
    int j = blockIdx.x * blockDim.x + threadIdx.x;
    if (j >= HID) return;
    float r = sigmoid_f(gi[j]           + gh[j]);
    float z = sigmoid_f(gi[j + HID]     + gh[j + HID]);
    float n = tanhf    (gi[j + 2 * HID] + r * gh[j + 2 * HID]);
    float h = (1.0f - z) * n + z * hidden[j];
    h_out[j]      = h;
    oc[NCAT + j]  = h;
}

// Single-block log-softmax over OUTN logits (3 strided passes over ~200KB).
__global__ __launch_bounds__(1024) void log_softmax_kernel(
    const float* __restrict__ logits, float* __restrict__ outp)
{
    __shared__ float red[32];
    __shared__ float s_max, s_logsum;
    const int tid = threadIdx.x;

    float m = -INFINITY;
    for (int i = tid; i < OUTN; i += 1024) m = fmaxf(m, logits[i]);
#pragma unroll
    for (int off = 16; off > 0; off >>= 1) m = fmaxf(m, __shfl_xor(m, off, 32));
    if ((tid & 31) == 0) red[tid >> 5] = m;
    __syncthreads();
    if (tid < 32) {
        float v = red[tid];
#pragma unroll
        for (int off = 16; off > 0; off >>= 1) v = fmaxf(v, __shfl_xor(v, off, 32));
        if (tid == 0) s_max = v;
    }
    __syncthreads();
    const float mm = s_max;

    float s = 0.0f;
    for (int i = tid; i < OUTN; i += 1024) s += __expf(logits[i] - mm);
#pragma unroll
    for (int off = 16; off > 0; off >>= 1) s += __shfl_xor(s, off, 32);
    __syncthreads();                 // red[] reuse
    if ((tid & 31) == 0) red[tid >> 5] = s;
    __syncthreads();
    if (tid < 32) {
        float v = red[tid];
#pragma unroll
        for (int off = 16; off > 0; off >>= 1) v += __shfl_xor(v, off, 32);
        if (tid == 0) s_logsum = logf(v);
    }
    __syncthreads();
    const float ls = s_logsum;

    for (int i = tid; i < OUTN; i += 1024) outp[i] = logits[i] - mm - ls;
}

extern "C" void kernel_launch(void* const* d_in, const int* in_sizes, int n_in,
                              void* d_out, int out_size, void* d_ws, size_t ws_size,
                              hipStream_t stream) {
    (void)in_sizes; (void)n_in; (void)out_size; (void)ws_size;
    const float* category = (const float*)d_in[0];
    const float* input    = (const float*)d_in[1];
    const float* hidden   = (const float*)d_in[2];
    const float* w_ih     = (const float*)d_in[3];
    const float* b_ih     = (const float*)d_in[4];
    const float* w_hh     = (const float*)d_in[5];
    const float* b_hh     = (const float*)d_in[6];
    const float* w_out    = (const float*)d_in[7];
    const float* b_out    = (const float*)d_in[8];

    float* logp  = (float*)d_out;          // [OUTN]
    float* h_out = logp + OUTN;            // [HID]

    float* ws     = (float*)d_ws;
    float* x      = ws;                    // 5120
    float* oc     = ws + 5120;             // 5120
    float* gi     = ws + 10240;            // 12288
    float* gh     = ws + 22528;            // 12288
    float* logits = ws + 34816;            // 50257

    // 1. x = [category, input]; oc[0:NCAT] = category
    build_x_kernel<<<(KXI + 255) / 256, 256, 0, stream>>>(category, input, x, oc);

    // 2. gi = x @ w_ih^T + b_ih          (12288 rows, K=5120 -> 768 waves)
    wmma_matvec_kernel<<<96, 256, 0, stream>>>(w_ih, x, b_ih, gi, 3 * HID, KXI);

    // 3. gh = hidden @ w_hh^T + b_hh     (12288 rows, K=4096)
    wmma_matvec_kernel<<<96, 256, 0, stream>>>(w_hh, hidden, b_hh, gh, 3 * HID, HID);

    // 4. gates -> h_new (also fills oc[NCAT:])
    gru_gate_kernel<<<(HID + 255) / 256, 256, 0, stream>>>(gi, gh, hidden, h_out, oc);

    // 5. logits = oc @ w_out^T + b_out   (50257 rows -> 3142 waves -> 393 blocks)
    wmma_matvec_kernel<<<393, 256, 0, stream>>>(w_out, oc, b_out, logits, OUTN, KXI);

    // 6. logp = log_softmax(logits)
    log_softmax_kernel<<<1, 1024, 0, stream>>>(logits, logp);
}